// AtnScore_12472585027765
// MI455X (gfx1250) — compile-verified
//
#include <hip/hip_runtime.h>
#include <hip/hip_bf16.h>
#include <math.h>

typedef float v2f __attribute__((ext_vector_type(2)));
typedef float v8f __attribute__((ext_vector_type(8)));

#define Bb   4
#define Cc   256
#define Nn   4096
#define BN   (Bb * Nn)          // 16384
#define SCALE 10.0f
#define EPS   1e-4f
#define CLMP  1e-8f
#define NCH   16                // softmax row-chunks
#define CHL   (Nn / NCH)        // 256

// ---------------------------------------------------------------------------
// K1: per-location inverse L2 norm over channels + mask -> coef, mm
// ---------------------------------------------------------------------------
__global__ __launch_bounds__(256) void prep_kernel(
    const float* __restrict__ x2, const float* __restrict__ mask,
    float* __restrict__ coef, float* __restrict__ mmv) {
  __shared__ float red[256];
  int t = threadIdx.x;
  int lg = t & 63;
  int cbase = (t >> 6) * 64;
  int gl = blockIdx.x * 64 + lg;            // [0, BN)
  int b = gl >> 12;
  int l = gl & (Nn - 1);
  const float* xp = x2 + (size_t)b * Cc * Nn + l;
  float s = 0.f;
#pragma unroll 8
  for (int c = 0; c < 64; ++c) {
    float v = xp[(size_t)(cbase + c) * Nn];
    s += v * v;
  }
  red[t] = s;
  __syncthreads();
  if (t < 64) {
    s = red[t] + red[t + 64] + red[t + 128] + red[t + 192];
    float inv = 1.0f / fmaxf(sqrtf(s), EPS);
    float m = mask[gl];
    float mmf = (m == 0.0f) ? 1.0f : 0.0f;
    coef[gl] = SCALE * mmf * inv;
    mmv[gl] = mmf;
  }
}

// ---------------------------------------------------------------------------
// K2: scaled Gram matrix  out[b][l][p] = coef[b][l] * sum_c X[b][c][l]*X[b][c][p]
// 128x128 macro tile / 256-thread block (8 waves, each 32x64), K-panels of 32.
// Both tiles staged in LDS transposed ([pos][c], stride 34) so every WMMA
// fragment (A and B) is one aligned ds_load_b64 into an even VGPR pair.
// ---------------------------------------------------------------------------
#define BM  128
#define BKc 32
#define LDT 34     // LDS tile stride: conflict-free + float2-aligned

__global__ __launch_bounds__(256) void gram_kernel(
    const float* __restrict__ X, const float* __restrict__ coef,
    float* __restrict__ out) {
  __shared__ float As[BM * LDT];    // As[l][c] : rows of output tile
  __shared__ float Bt[BM * LDT];    // Bt[p][c] : cols of output tile (transposed)

  int t    = threadIdx.x;
  int wave = t >> 5;
  int lane = t & 31;
  int b    = blockIdx.z;
  int L0   = blockIdx.y * BM;
  int P0   = blockIdx.x * BM;
  const float* Xb = X + (size_t)b * Cc * Nn;

  int wrow = wave & 3;              // wave's 32-row group
  int wcol = wave >> 2;             // wave's 64-col group
  int mrow  = lane & 15;            // M (A) / N (B) within 16x16 sub-tile
  int khalf = (lane >> 4) * 2;      // K-pair base per lane half

  v8f acc[2][4] = {};               // 2 row-subtiles x 4 col-subtiles

  for (int kt = 0; kt < Cc; kt += BKc) {
    // Stage A^T: As[l][c] = X[kt+c][L0+l]   (coalesced in l)
#pragma unroll
    for (int i = 0; i < 16; ++i) {
      int idx = t + 256 * i;
      int c = idx >> 7;
      int l = idx & 127;
      As[l * LDT + c] = Xb[(size_t)(kt + c) * Nn + (L0 + l)];
    }
    // Stage B^T: Bt[p][c] = X[kt+c][P0+p]   (coalesced in p)
#pragma unroll
    for (int i = 0; i < 16; ++i) {
      int idx = t + 256 * i;
      int c = idx >> 7;
      int p = idx & 127;
      Bt[p * LDT + c] = Xb[(size_t)(kt + c) * Nn + (P0 + p)];
    }
    __syncthreads();

    // Prefetch next K-panel into caches while we compute this one.
    if (kt + BKc < Cc) {
      int j = t & 127;                       // 128 cache lines per 16KB tile
      int c = j >> 2;
      int off = (j & 3) * 32;
      const float* pf = (t < 128)
          ? &Xb[(size_t)(kt + BKc + c) * Nn + (L0 + off)]
          : &Xb[(size_t)(kt + BKc + c) * Nn + (P0 + off)];
      __builtin_prefetch(pf, 0, 0);
    }

#pragma unroll
    for (int k4 = 0; k4 < BKc; k4 += 4) {
      v2f a0 = *(const v2f*)&As[(wrow * 32 +      mrow) * LDT + k4 + khalf];
      v2f a1 = *(const v2f*)&As[(wrow * 32 + 16 + mrow) * LDT + k4 + khalf];
#pragma unroll
      for (int cb = 0; cb < 4; ++cb) {
        v2f bb = *(const v2f*)&Bt[(wcol * 64 + cb * 16 + mrow) * LDT + k4 + khalf];
        acc[0][cb] = __builtin_amdgcn_wmma_f32_16x16x4_f32(
            false, a0, false, bb, (short)0, acc[0][cb], false, false);
        acc[1][cb] = __builtin_amdgcn_wmma_f32_16x16x4_f32(
            false, a1, false, bb, (short)0, acc[1][cb], false, false);
      }
    }
    __syncthreads();
  }

  // Epilogue: row-scale by coef, store softmax argument
  size_t outb = (size_t)b * Nn * Nn;
#pragma unroll
  for (int mi = 0; mi < 2; ++mi) {
    int rowbase = L0 + wrow * 32 + mi * 16 + (lane >> 4) * 8;  // D: VGPR r -> M=r (+8 hi half)
    float cf[8];
#pragma unroll
    for (int r = 0; r < 8; ++r) cf[r] = coef[b * Nn + rowbase + r];
#pragma unroll
    for (int cb = 0; cb < 4; ++cb) {
      int col = P0 + wcol * 64 + cb * 16 + mrow;
#pragma unroll
      for (int r = 0; r < 8; ++r) {
        out[outb + (size_t)(rowbase + r) * Nn + col] = acc[mi][cb][r] * cf[r];
      }
    }
  }
}

// ---------------------------------------------------------------------------
// K3a: chunked online softmax stats over l (columns of out), coalesced in p
// ---------------------------------------------------------------------------
__global__ __launch_bounds__(256) void stats_kernel(
    const float* __restrict__ out, float* __restrict__ Mp, float* __restrict__ Sp) {
  int g = blockIdx.x * 256 + threadIdx.x;      // [0, NCH*BN)
  int q = g / BN;
  int rem = g - q * BN;                        // b*Nn + p
  const float* colp = out + (size_t)(rem >> 12) * Nn * Nn + (rem & (Nn - 1));
  float m = -INFINITY, s = 0.f;
  int l0 = q * CHL;
  for (int i = 0; i < CHL; ++i) {
    float x = colp[(size_t)(l0 + i) * Nn];
    float nm = fmaxf(m, x);
    s = s * __expf(m - nm) + __expf(x - nm);
    m = nm;
  }
  Mp[g] = m;
  Sp[g] = s;
}

// K3b: merge chunk stats (log-sum-exp combine)
__global__ __launch_bounds__(256) void combine_kernel(
    const float* __restrict__ Mp, const float* __restrict__ Sp,
    float* __restrict__ Mx, float* __restrict__ Rd) {
  int g = blockIdx.x * 256 + threadIdx.x;      // [0, BN)
  float m = -INFINITY, s = 0.f;
#pragma unroll
  for (int q = 0; q < NCH; ++q) {
    float mq = Mp[q * BN + g];
    float sq = Sp[q * BN + g];
    float nm = fmaxf(m, mq);
    s = s * __expf(m - nm) + sq * __expf(mq - nm);
    m = nm;
  }
  Mx[g] = m;
  Rd[g] = 1.0f / s;
}

// ---------------------------------------------------------------------------
// K3c: final  out = max( mm[l] * exp(x - M[b,p]) * R[b,p], 1e-8 ), float4
// ---------------------------------------------------------------------------
__global__ __launch_bounds__(256) void finalize_kernel(
    float* __restrict__ out, const float* __restrict__ mmv,
    const float* __restrict__ Mx, const float* __restrict__ Rd) {
  size_t g = (size_t)blockIdx.x * 256 + threadIdx.x;
  size_t flat = g * 4;
  int b = (int)(flat >> 24);                    // N*N = 2^24
  size_t r = flat & (((size_t)1 << 24) - 1);
  int l = (int)(r >> 12);
  int p = (int)(r & (Nn - 1));                  // multiple of 4
  float mmf = mmv[b * Nn + l];
  int bp = b * Nn + p;
  float4 x = *(float4*)&out[flat];
  float4 y;
  y.x = fmaxf(mmf * __expf(x.x - Mx[bp + 0]) * Rd[bp + 0], CLMP);
  y.y = fmaxf(mmf * __expf(x.y - Mx[bp + 1]) * Rd[bp + 1], CLMP);
  y.z = fmaxf(mmf * __expf(x.z - Mx[bp + 2]) * Rd[bp + 2], CLMP);
  y.w = fmaxf(mmf * __expf(x.w - Mx[bp + 3]) * Rd[bp + 3], CLMP);
  *(float4*)&out[flat] = y;
}

// ---------------------------------------------------------------------------
extern "C" void kernel_launch(void* const* d_in, const int* in_sizes, int n_in,
                              void* d_out, int out_size, void* d_ws, size_t ws_size,
                              hipStream_t stream) {
  (void)in_sizes; (void)n_in; (void)out_size; (void)ws_size;
  const float* x2   = (const float*)d_in[0];
  const float* mask = (const float*)d_in[1];
  float* out = (float*)d_out;

  // workspace layout (floats): coef[BN], mm[BN], Mp[NCH*BN], Sp[NCH*BN], Mx[BN], Rd[BN]
  float* ws   = (float*)d_ws;
  float* coef = ws;
  float* mmv  = ws + BN;
  float* Mp   = ws + 2 * BN;
  float* Sp   = ws + (2 + NCH) * BN;
  float* Mx   = ws + (2 + 2 * NCH) * BN;
  float* Rd   = ws + (3 + 2 * NCH) * BN;

  prep_kernel<<<BN / 64, 256, 0, stream>>>(x2, mask, coef, mmv);
  gram_kernel<<<dim3(Nn / BM, Nn / BM, Bb), 256, 0, stream>>>(x2, coef, out);
  stats_kernel<<<(NCH * BN) / 256, 256, 0, stream>>>(out, Mp, Sp);
  combine_kernel<<<BN / 256, 256, 0, stream>>>(Mp, Sp, Mx, Rd);
  finalize_kernel<<<((size_t)Bb * Nn * Nn) / (4 * 256), 256, 0, stream>>>(out, mmv, Mx, Rd);
}